// LogisticModel_83760452207495
// MI455X (gfx1250) — compile-verified
//
#include <hip/hip_runtime.h>
#include <hip/hip_bf16.h>
#include <math.h>

#define DIM   512
#define BATCH 1024
#define NPAIR (DIM * (DIM - 1) / 2)      // 130816
#define W_PAIR_OFF DIM                   // 512
#define W_SQ_OFF   (DIM + NPAIR)         // 131328

typedef __attribute__((ext_vector_type(2))) float v2f;
typedef __attribute__((ext_vector_type(8))) float v8f;

// Upper-triangular quadratic coefficient matrix M (from W):
//   M[k][j] = W_pair(k,j) for k<j  (row-major triu order, matches np.triu_indices)
//   M[k][k] = W_sq[k]
//   M[k][j] = 0 for k>j
__device__ __forceinline__ float coefM(int k, int j, const float* __restrict__ W) {
  if (k < j) {
    const int idx = W_PAIR_OFF + k * (DIM - 1) - ((k * (k - 1)) >> 1) + (j - k - 1);
    return W[idx];
  }
  if (k == j) return W[W_SQ_OFF + k];
  return 0.0f;
}

// ---------------------------------------------------------------------------
// Kernel 1: build M in k-pair-packed layout (float2 per (pair,col)):
//   Mp2[p*512 + j] = ( M[2p][j], M[2p+1][j] )
// so a WMMA B-fragment lane (pair p = k/2 + half, col) is one b64 load.
// Size: 512*512*4 B = 1 MB in d_ws (L2-resident thereafter).
// ---------------------------------------------------------------------------
__global__ __launch_bounds__(256)
void build_M_packed(const float* __restrict__ W, float* __restrict__ Mp) {
  const int idx = blockIdx.x * 256 + threadIdx.x;  // 0 .. 512*512-1
  const int k = idx >> 9;                          // M row
  const int j = idx & (DIM - 1);                   // M col
  const float v = coefM(k, j, W);
  Mp[((((k >> 1) << 9) | j) << 1) | (k & 1)] = v;
}

// ---------------------------------------------------------------------------
// Kernel 2: fused  out = sigmoid( X·W_lin + rowdot(X·M, X) + b )
// One wave per 16 batch rows; 4 waves per block split the 32 N-tiles.
// Inner loop: 4x global_load_b64 + 2x v_wmma_f32_16x16x4_f32 on two
// independent accumulator chains (breaks the WMMA RAW chain, hides load lat).
// ---------------------------------------------------------------------------
__global__ __launch_bounds__(128)
void poly2_logistic_fused(const float* __restrict__ X,
                          const float* __restrict__ W,
                          const float* __restrict__ bias,
                          const float* __restrict__ Mp,
                          float* __restrict__ out)
{
  const int lane = threadIdx.x & 31;   // wave32 lane
  const int wv   = threadIdx.x >> 5;   // wave id in block: 0..3
  const int half = lane >> 4;          // 0 | 1 (half-wave)
  const int l15  = lane & 15;
  const int m0   = blockIdx.x * 16;    // batch-row base of this block

  __shared__ float sq[4][16];          // per-wave quadratic partials per row
  __shared__ float slin[16];           // linear term per row

  // ---------------- linear term x . W_lin (wave 0 only) ----------------
  if (wv == 0) {
    const int row = m0 + l15;
    const float4* xr = reinterpret_cast<const float4*>(X + row * DIM);
    const float4* wr = reinterpret_cast<const float4*>(W);
    float lin = 0.0f;
    #pragma unroll 4
    for (int q = half * (DIM / 8); q < (half + 1) * (DIM / 8); ++q) {
      const float4 xv = xr[q];
      const float4 wl = wr[q];
      lin += xv.x * wl.x + xv.y * wl.y + xv.z * wl.z + xv.w * wl.w;
    }
    lin += __shfl_xor(lin, 16);        // combine the two half-wave partials
    if (lane < 16) slin[lane] = lin;
  }

  // ---------------- quadratic term x^T M x via f32 WMMA ----------------
  float pacc[8];
  #pragma unroll
  for (int r = 0; r < 8; ++r) pacc[r] = 0.0f;

  const v2f* xrow2 = reinterpret_cast<const v2f*>(X + (m0 + l15) * DIM);
  const v2f* Mp2   = reinterpret_cast<const v2f*>(Mp);

  for (int jt = wv; jt < DIM / 16; jt += 4) { // waves split the N tiles
    const int j0   = jt * 16;
    const int col  = j0 + l15;
    const int kmax = j0 + 16;                 // M is upper-triangular: k <= j
    v8f c0 = {0.f, 0.f, 0.f, 0.f, 0.f, 0.f, 0.f, 0.f};
    v8f c1 = {0.f, 0.f, 0.f, 0.f, 0.f, 0.f, 0.f, 0.f};
    // kmax = 16*(jt+1): always a multiple of 8 -> exact 2-way unroll
    for (int k = 0; k < kmax; k += 8) {
      const int p0 = (k >> 1) + half;         // k-pair index: 2p = kk = k + 2*half
      const int p1 = p0 + 2;
      const v2f a0 = xrow2[p0];               // ( x[kk], x[kk+1] )
      const v2f b0 = Mp2[(p0 << 9) | col];    // ( M[kk][col], M[kk+1][col] )
      const v2f a1 = xrow2[p1];
      const v2f b1 = Mp2[(p1 << 9) | col];
      c0 = __builtin_amdgcn_wmma_f32_16x16x4_f32(
              false, a0, false, b0, (short)0, c0, false, false);
      c1 = __builtin_amdgcn_wmma_f32_16x16x4_f32(
              false, a1, false, b1, (short)0, c1, false, false);
    }
    // Epilogue: C/D layout -> VGPR r, lane<16: (M=r, N=l15); lane>=16: (M=8+r, N=l15)
    #pragma unroll
    for (int r = 0; r < 8; ++r) {
      pacc[r] += (c0[r] + c1[r]) * X[(m0 + half * 8 + r) * DIM + col];
    }
  }

  // Reduce each pacc[r] across the 16 lanes of this half-wave (masks stay in-half)
  #pragma unroll
  for (int r = 0; r < 8; ++r) {
    float v = pacc[r];
    v += __shfl_xor(v, 1);
    v += __shfl_xor(v, 2);
    v += __shfl_xor(v, 4);
    v += __shfl_xor(v, 8);
    pacc[r] = v;
  }
  if (l15 == 0) {
    #pragma unroll
    for (int r = 0; r < 8; ++r) sq[wv][half * 8 + r] = pacc[r];
  }

  __syncthreads();

  // ---------------- combine, bias, sigmoid ----------------
  if (threadIdx.x < 16) {
    const float z = sq[0][threadIdx.x] + sq[1][threadIdx.x]
                  + sq[2][threadIdx.x] + sq[3][threadIdx.x]
                  + slin[threadIdx.x] + bias[0];
    out[m0 + threadIdx.x] = 1.0f / (1.0f + __expf(-z));
  }
}

extern "C" void kernel_launch(void* const* d_in, const int* in_sizes, int n_in,
                              void* d_out, int out_size, void* d_ws, size_t ws_size,
                              hipStream_t stream) {
  (void)in_sizes; (void)n_in; (void)ws_size; (void)out_size;
  const float* X = (const float*)d_in[0];   // [1024, 512] f32
  const float* W = (const float*)d_in[1];   // [1, 131840] f32
  const float* B = (const float*)d_in[2];   // [1] f32
  float* out = (float*)d_out;               // [1024] f32 (sigmoid probs)
  float* Mp  = (float*)d_ws;                // 512*512*4 = 1 MB packed coefficient matrix

  build_M_packed<<<dim3((DIM * DIM) / 256), dim3(256), 0, stream>>>(W, Mp);
  poly2_logistic_fused<<<dim3(BATCH / 16), dim3(128), 0, stream>>>(X, W, B, Mp, out);
}